// DifferentiableModalPlate_833223655862
// MI455X (gfx1250) — compile-verified
//
#include <hip/hip_runtime.h>
#include <math.h>

typedef __attribute__((ext_vector_type(2))) float v2f;
typedef __attribute__((ext_vector_type(8))) float v8f;

#define KDT      (1.0f / 44100.0f)
#define PI_F     3.14159265358979323846f
#define MAX_OM_F (10000.0f * 2.0f * PI_F)
#define MIN_OM_F (20.0f * 2.0f * PI_F)
#define LX_F     0.5f
// ALPHA = 3*ln10/TAU0 ; BETA = 3*ln10*(1/TAU1-1/TAU0)/(2*pi*500)^2  (doubles -> f32, as numpy does)
#define ALPHA_F  ((float)(3.0 * 2.302585092994046 / 6.0))
#define BETA_F   ((float)(3.0 * 2.302585092994046 * (1.0 - 1.0/6.0) / \
                          ((2.0 * 3.14159265358979323846 * 500.0) * (2.0 * 3.14159265358979323846 * 500.0))))

#define NMODES   6400
#define KTOT     (2 * NMODES)      // 12800 GEMM K-dim
#define NCHUNK   (NMODES / 2)      // 3200 WMMA K=4 chunks (2 modes each)

static __device__ __forceinline__ float softplusf(float x) {
    return (x > 20.0f) ? x : log1pf(__expf(x));
}

// ---------- kernel 1: per-mode parameters (sigma, omega*K, P/sin(omega*K)) ----------
__global__ void k_modes(const float* mu_raw, const float* Dmu_raw, const float* T0mu_raw,
                        const float* Ly_raw, const float* xo_raw, const float* yo_raw,
                        float* __restrict__ sigma_o, float* __restrict__ wK_o,
                        float* __restrict__ Psc_o) {
    int m = blockIdx.x * blockDim.x + threadIdx.x;
    if (m >= NMODES) return;
    float mu   = softplusf(mu_raw[0])   + 1e-4f;
    float Dmu  = softplusf(Dmu_raw[0])  + 1e-4f;
    float T0mu = softplusf(T0mu_raw[0]) + 1e-4f;
    float Ly = 1.1f + (4.0f - 1.1f) * ((tanhf(Ly_raw[0]) + 1.0f) * 0.5f);
    float xo = 0.49f * LX_F + (1.0f - 0.49f) * LX_F * ((tanhf(xo_raw[0]) + 1.0f) * 0.5f);
    float yo = 0.51f * Ly   + (1.0f - 0.51f) * Ly   * ((tanhf(yo_raw[0]) + 1.0f) * 0.5f);
    float xi = 0.1f * LX_F;
    float yi = 0.1f * Ly;

    float M  = (float)(m / 80 + 1);
    float Nn = (float)(m % 80 + 1);
    float gm = M * PI_F / LX_F;
    float gn = Nn * PI_F / Ly;
    float g1 = gm * gm + gn * gn;
    float om2 = T0mu * g1 + Dmu * g1 * g1;
    float omega = sqrtf(fmaxf(om2, 0.0f));
    float valid = (omega <= MAX_OM_F && omega >= MIN_OM_F) ? 1.0f : 0.0f;

    float in_w  = __cosf(xi * PI_F * M / LX_F) * __cosf(yi * PI_F * Nn / Ly);
    float out_w = __cosf(xo * PI_F * M / LX_F) * __cosf(yo * PI_F * Nn / Ly);
    float sig = ALPHA_F + BETA_F * omega * omega;
    float ms  = 0.25f * mu * LX_F * Ly;
    float P   = out_w * in_w * (KDT * KDT) * __expf(-sig * KDT) / ms * valid;
    float inv_sin = 1.0f / (__sinf(omega * KDT) + 1e-8f);

    sigma_o[m] = sig;
    wK_o[m]    = omega * KDT;
    Psc_o[m]   = P * inv_sin;
}

// ---------- kernel 2: B matrix [12800 x 16]: row 2m = P'*e_r*cos(r*wK), row 2m+1 = P'*e_r*sin(r*wK)
__global__ void k_bmat(const float* __restrict__ sigma, const float* __restrict__ wK,
                       const float* __restrict__ Psc, float* __restrict__ B) {
    int idx = blockIdx.x * blockDim.x + threadIdx.x;
    if (idx >= NMODES * 16) return;
    int m = idx >> 4;
    int r = idx & 15;
    float sg = sigma[m];
    float wk = wK[m];
    float base = Psc[m] * __expf(-sg * (float)(r - 1) * KDT);   // decay uses (n-1)
    float s, c;
    __sincosf((float)r * wk, &s, &c);
    B[(2 * m) * 16 + r]     = base * c;
    B[(2 * m + 1) * 16 + r] = base * s;
}

// ---------- kernel 3: main GEMM via V_WMMA_F32_16X16X4_F32 ----------
// One block = one 256-sample tile. 8 waves split the 3200 K-chunks, LDS reduce.
__global__ void __launch_bounds__(256) k_synth(const float* __restrict__ sigma,
                                               const float* __restrict__ wK,
                                               const float* __restrict__ B,
                                               float* __restrict__ unnorm, int nsamp) {
    __shared__ float red[8][256];
    const int lane = threadIdx.x & 31;
    const int wave = threadIdx.x >> 5;
    const int q    = lane & 15;       // A row (M) for lanes' half, also B col (N)
    const int hi   = lane >> 4;       // 0: K=0/1 pair, 1: K=2/3 pair
    const int qg   = blockIdx.x * 16 + q;          // global q index (sample block)
    const float t16 = 16.0f * (float)qg;           // in sample units; time = t16*K

    v8f acc = {};
    for (int ch = wave; ch < NCHUNK; ch += 8) {
        int m = 2 * ch + hi;                       // mode handled by this lane
        float sg = sigma[m];
        float wk = wK[m];
        float E = __expf(-sg * (t16 * KDT));
        float S, C;
        __sincosf(t16 * wk, &S, &C);
        v2f a;
        a.x = E * S;                               // A[q, k even] = E*sin
        a.y = E * C;                               // A[q, k odd]  = E*cos
        const float* Bp = B + (4 * ch + 2 * hi) * 16 + q;
        v2f b;
        b.x = Bp[0];                               // B[k even, r]
        b.y = Bp[16];                              // B[k odd,  r]
        acc = __builtin_amdgcn_wmma_f32_16x16x4_f32(false, a, false, b,
                                                    (short)0, acc, false, false);
    }
    // D layout: acc[j] -> M = j + 8*hi, N = q ; flat in-tile sample = M*16 + N
#pragma unroll
    for (int j = 0; j < 8; ++j)
        red[wave][(j + 8 * hi) * 16 + q] = acc[j];
    __syncthreads();

    int t = threadIdx.x;
    float s = 0.0f;
#pragma unroll
    for (int w = 0; w < 8; ++w) s += red[w][t];
    int n = blockIdx.x * 256 + t;
    if (n < nsamp) unnorm[n] = s;
}

// ---------- kernel 4: single-block max|x| ----------
__global__ void k_max(const float* __restrict__ unnorm, float* __restrict__ maxval, int nsamp) {
    __shared__ float sm[256];
    float m = 0.0f;
    for (int i = threadIdx.x; i < nsamp; i += 256) m = fmaxf(m, fabsf(unnorm[i]));
    sm[threadIdx.x] = m;
    __syncthreads();
    for (int s = 128; s > 0; s >>= 1) {
        if (threadIdx.x < s) sm[threadIdx.x] = fmaxf(sm[threadIdx.x], sm[threadIdx.x + s]);
        __syncthreads();
    }
    if (threadIdx.x == 0) maxval[0] = sm[0];
}

// ---------- kernel 5: normalize ----------
__global__ void k_norm(const float* __restrict__ unnorm, const float* __restrict__ maxval,
                       float* __restrict__ out, int nsamp) {
    int i = blockIdx.x * blockDim.x + threadIdx.x;
    if (i < nsamp) out[i] = unnorm[i] / (maxval[0] + 1e-8f);
}

extern "C" void kernel_launch(void* const* d_in, const int* in_sizes, int n_in,
                              void* d_out, int out_size, void* d_ws, size_t ws_size,
                              hipStream_t stream) {
    const float* mu_raw   = (const float*)d_in[0];
    const float* Dmu_raw  = (const float*)d_in[1];
    const float* T0mu_raw = (const float*)d_in[2];
    const float* Ly_raw   = (const float*)d_in[3];
    const float* xo_raw   = (const float*)d_in[4];
    const float* yo_raw   = (const float*)d_in[5];
    // d_in[6] is num_samples on device; out_size equals it on host.
    const int nsamp = out_size;

    float* w      = (float*)d_ws;
    float* sigma  = w;                       // 6400
    float* wk     = w + NMODES;              // 6400
    float* psc    = w + 2 * NMODES;          // 6400
    float* B      = w + 3 * NMODES;          // 12800*16 = 204800
    float* unnorm = B + KTOT * 16;           // nsamp
    float* maxval = unnorm + nsamp;          // 1

    k_modes<<<(NMODES + 255) / 256, 256, 0, stream>>>(mu_raw, Dmu_raw, T0mu_raw,
                                                      Ly_raw, xo_raw, yo_raw,
                                                      sigma, wk, psc);
    k_bmat<<<(NMODES * 16 + 255) / 256, 256, 0, stream>>>(sigma, wk, psc, B);

    int tiles = (nsamp + 255) / 256;
    k_synth<<<tiles, 256, 0, stream>>>(sigma, wk, B, unnorm, nsamp);

    k_max<<<1, 256, 0, stream>>>(unnorm, maxval, nsamp);
    k_norm<<<(nsamp + 255) / 256, 256, 0, stream>>>(unnorm, maxval, (float*)d_out, nsamp);
}